// SAN_SAKT_50998441673207
// MI455X (gfx1250) — compile-verified
//
#include <hip/hip_runtime.h>
#include <math.h>

// ---------------------------------------------------------------------------
// CDNA5 (gfx1250) fp32 WMMA helpers: V_WMMA_F32_16X16X4_F32, wave32.
//
// A 16x4 fp32 layout (ISA 7.12.2): lanes 0-15 hold M=lane, v0=K0, v1=K1;
//                                  lanes 16-31 hold M=lane-16, v0=K2, v1=K3.
// B 4x16 fp32 layout (row-striped like C/D): lanes 0-15 -> K=0(v0),K=1(v1);
//                                            lanes 16-31 -> K=2(v0),K=3(v1).
// C/D 16x16 fp32: VGPR r, lanes 0-15 -> D[r, lane]; lanes 16-31 -> D[r+8, lane-16].
// ---------------------------------------------------------------------------

typedef __attribute__((ext_vector_type(2))) float v2f;
typedef __attribute__((ext_vector_type(8))) float v8f;

#define SEQ 512
#define DM  256
#define NH  8
#define HD  32
#define NROW 16384   // B*S

__device__ __forceinline__ v8f wmma_acc_tile(const float* A, int lda,
                                             const float* B, int ldb,
                                             int K, v8f acc) {
  const int lane = threadIdx.x & 31;
  const int half = lane >> 4;
  const int l    = lane & 15;
  const float* Ap = A + (size_t)l * lda + 2 * half;
  const float* Bp = B + (size_t)(2 * half) * ldb + l;
#pragma unroll 4
  for (int k = 0; k < K; k += 4) {
    v2f a, b;
    a[0] = Ap[k];
    a[1] = Ap[k + 1];
    b[0] = Bp[(size_t)k * ldb];
    b[1] = Bp[(size_t)(k + 1) * ldb];
    acc = __builtin_amdgcn_wmma_f32_16x16x4_f32(false, a, false, b,
                                                (short)0, acc, false, false);
  }
  return acc;
}

__device__ __forceinline__ void wmma_store_tile(float* C, int ldc, v8f c) {
  const int lane = threadIdx.x & 31;
  const int half = lane >> 4;
  const int l    = lane & 15;
#pragma unroll
  for (int r = 0; r < 8; ++r)
    C[(size_t)(r + 8 * half) * ldc + l] = c[r];
}

// ---------------------------------------------------------------------------
// Kernel: transpose pos_key_emb [512,32] -> [32,512] so q@pke^T is row-major.
// ---------------------------------------------------------------------------
__global__ void k_transpose_pke(const float* __restrict__ pke,
                                float* __restrict__ pkeT) {
  int idx = blockIdx.x * blockDim.x + threadIdx.x;
  if (idx < SEQ * HD) {
    int r = idx / HD, d = idx % HD;
    pkeT[(size_t)d * SEQ + r] = pke[idx];
  }
}

// ---------------------------------------------------------------------------
// Kernel: inputs = relu(interact @ lin_in_w + b) + cosine_pe(s)
// A:[16384,512] @ B:[512,256]. One wave per 16x16 tile.
// ---------------------------------------------------------------------------
__global__ __launch_bounds__(32)
void k_lin_in(const float* __restrict__ X, const float* __restrict__ W,
              const float* __restrict__ bias, float* __restrict__ Y) {
  const int tile = blockIdx.x;                 // 1024 * 16 tiles
  const int m0 = (tile >> 4) * 16;
  const int n0 = (tile & 15) * 16;
  v8f acc = {};
  acc = wmma_acc_tile(X + (size_t)m0 * (2 * DM), 2 * DM, W + n0, DM, 2 * DM, acc);

  const int lane = threadIdx.x & 31, half = lane >> 4, l = lane & 15;
  const float ln1e4_over_D = 9.210340371976184f / (float)DM;  // ln(10000)/256
#pragma unroll
  for (int r = 0; r < 8; ++r) {
    int row = m0 + r + 8 * half;
    int col = n0 + l;
    float v = fmaxf(acc[r] + bias[col], 0.0f);
    int s = row & (SEQ - 1);
    int i = col >> 1;
    float div = expf(-(float)(2 * i) * ln1e4_over_D);
    float ang = (float)s * div;
    float pe = (col & 1) ? cosf(ang) : sinf(ang);
    Y[(size_t)row * DM + col] = v + pe;
  }
}

// ---------------------------------------------------------------------------
// Kernel: QKV projections, scattering to head layouts.
// q -> [B,H,S,HD], k -> [B,H,HD,S] (transposed), v -> [B,H,S,HD].
// blockIdx.y selects q/k/v.
// ---------------------------------------------------------------------------
__global__ __launch_bounds__(32)
void k_qkv(const float* __restrict__ Qin, const float* __restrict__ Xin,
           const float* __restrict__ Wq, const float* __restrict__ bq,
           const float* __restrict__ Wk, const float* __restrict__ bk,
           const float* __restrict__ Wv, const float* __restrict__ bv,
           float* __restrict__ qb, float* __restrict__ kTb,
           float* __restrict__ vb) {
  const int which = blockIdx.y;                // 0=q 1=k 2=v
  const int tile = blockIdx.x;                 // 1024 * 16
  const int m0 = (tile >> 4) * 16;
  const int n0 = (tile & 15) * 16;

  const float* A    = (which == 0) ? Qin : Xin;
  const float* W    = (which == 0) ? Wq : (which == 1) ? Wk : Wv;
  const float* bias = (which == 0) ? bq : (which == 1) ? bk : bv;

  v8f acc = {};
  acc = wmma_acc_tile(A + (size_t)m0 * DM, DM, W + n0, DM, DM, acc);

  const int lane = threadIdx.x & 31, half = lane >> 4, l = lane & 15;
#pragma unroll
  for (int r = 0; r < 8; ++r) {
    int row = m0 + r + 8 * half;               // b*S + s
    int col = n0 + l;                          // d_model
    int b = row >> 9, s = row & (SEQ - 1);
    int h = col >> 5, hd = col & (HD - 1);
    float val = acc[r] + bias[col];
    size_t bh = (size_t)(b * NH + h);
    if (which == 1)      kTb[(bh * HD + hd) * SEQ + s] = val;
    else if (which == 0) qb [(bh * SEQ + s) * HD + hd] = val;
    else                 vb [(bh * SEQ + s) * HD + hd] = val;
  }
}

// ---------------------------------------------------------------------------
// Kernel: relative-position causal attention. One wave per (b, h, 16-row tile).
//  S1 = q@K^T                       (WMMA, K=32)
//  R  = q@pke^T  (Toeplitz GEMM)    (WMMA, K=32)
//  scores = (S1[i,j] + R[i,i-j]) / sqrt(32), j>i -> -1e9; softmax -> p
//  p'[i,r] = p[i,i-r]
//  out = p@V + p'@pve               (WMMA, K=512, shared accumulator)
// ---------------------------------------------------------------------------
#define SPAD 513
__global__ __launch_bounds__(32)
void k_attn(const float* __restrict__ qb, const float* __restrict__ kTb,
            const float* __restrict__ vb, const float* __restrict__ pkeT,
            const float* __restrict__ pve, float* __restrict__ outb) {
  __shared__ float Sbuf[16][SPAD];   // scores -> p
  __shared__ float Rbuf[16][SPAD];   // rel scores -> p'

  const int blk = blockIdx.x;        // B*H*(S/16) = 8192
  const int it  = blk & 31;
  const int bh  = blk >> 5;
  const int h   = bh & (NH - 1);
  const int b   = bh >> 3;
  const int i0  = it * 16;

  const float* qtile = qb + ((size_t)bh * SEQ + i0) * HD;   // [16,32], lda=32
  const float* Kt    = kTb + (size_t)bh * HD * SEQ;         // [32,512]

  // 1) content scores q @ K^T  (32 column tiles of 16)
  for (int jt = 0; jt < SEQ / 16; ++jt) {
    v8f acc = {};
    acc = wmma_acc_tile(qtile, HD, Kt + jt * 16, SEQ, HD, acc);
    wmma_store_tile(&Sbuf[0][0] + jt * 16, SPAD, acc);
  }
  // 2) relative scores R[i,r] = q[i,:] . pke[r,:]
  for (int jt = 0; jt < SEQ / 16; ++jt) {
    v8f acc = {};
    acc = wmma_acc_tile(qtile, HD, pkeT + jt * 16, SEQ, HD, acc);
    wmma_store_tile(&Rbuf[0][0] + jt * 16, SPAD, acc);
  }
  __syncthreads();

  // 3) combine (Toeplitz gather), scale, causal mask
  const float inv_s = 0.17677669529663687f;   // 1/sqrt(32)
  for (int r = 0; r < 16; ++r) {
    int i = i0 + r;
    for (int j = (int)threadIdx.x; j < SEQ; j += 32) {
      float sc = (j <= i) ? (Sbuf[r][j] + Rbuf[r][i - j]) * inv_s : -1e9f;
      Sbuf[r][j] = sc;
    }
  }
  __syncthreads();

  // 4) row softmax (wave32 shuffle reductions)
  for (int r = 0; r < 16; ++r) {
    float m = -3.0e38f;
    for (int j = (int)threadIdx.x; j < SEQ; j += 32) m = fmaxf(m, Sbuf[r][j]);
#pragma unroll
    for (int off = 16; off; off >>= 1) m = fmaxf(m, __shfl_xor(m, off, 32));
    float sum = 0.f;
    for (int j = (int)threadIdx.x; j < SEQ; j += 32) {
      float e = expf(Sbuf[r][j] - m);
      Sbuf[r][j] = e;
      sum += e;
    }
#pragma unroll
    for (int off = 16; off; off >>= 1) sum += __shfl_xor(sum, off, 32);
    float inv = 1.0f / sum;
    for (int j = (int)threadIdx.x; j < SEQ; j += 32) Sbuf[r][j] *= inv;
  }
  __syncthreads();

  // 5) reversed probs p'[i,r] = p[i, i-r]  (r > i contributes 0)
  for (int r = 0; r < 16; ++r) {
    int i = i0 + r;
    for (int rr = (int)threadIdx.x; rr < SEQ; rr += 32)
      Rbuf[r][rr] = (rr <= i) ? Sbuf[r][i - rr] : 0.0f;
  }
  __syncthreads();

  // 6) out = p @ V + p' @ pve   (two 16-wide column tiles of HD=32)
  const float* V = vb + (size_t)bh * SEQ * HD;
  const int lane = threadIdx.x & 31, half = lane >> 4, l = lane & 15;
  for (int nt = 0; nt < 2; ++nt) {
    v8f acc = {};
    acc = wmma_acc_tile(&Sbuf[0][0], SPAD, V + nt * 16, HD, SEQ, acc);
    acc = wmma_acc_tile(&Rbuf[0][0], SPAD, pve + nt * 16, HD, SEQ, acc);
#pragma unroll
    for (int r = 0; r < 8; ++r) {
      int s = i0 + r + 8 * half;
      int d = h * HD + nt * 16 + l;
      outb[((size_t)b * SEQ + s) * DM + d] = acc[r];
    }
  }
}

// ---------------------------------------------------------------------------
// Kernel: generic WMMA GEMM with fused bias / relu / residual epilogue.
// ---------------------------------------------------------------------------
__global__ __launch_bounds__(32)
void k_gemm(const float* __restrict__ A, int lda,
            const float* __restrict__ B, int ldb,
            const float* __restrict__ bias, const float* __restrict__ resid,
            float* __restrict__ C, int ldc, int K, int nTilesN, int do_relu) {
  const int tile = blockIdx.x;
  const int m0 = (tile / nTilesN) * 16;
  const int n0 = (tile % nTilesN) * 16;
  v8f acc = {};
  acc = wmma_acc_tile(A + (size_t)m0 * lda, lda, B + n0, ldb, K, acc);
  const int lane = threadIdx.x & 31, half = lane >> 4, l = lane & 15;
#pragma unroll
  for (int r = 0; r < 8; ++r) {
    int row = m0 + r + 8 * half;
    int col = n0 + l;
    float v = acc[r] + bias[col];
    if (do_relu) v = fmaxf(v, 0.0f);
    if (resid)   v += resid[(size_t)row * ldc + col];
    C[(size_t)row * ldc + col] = v;
  }
}

// ---------------------------------------------------------------------------
// Kernel: layer norm over D=256 per row; optional pre-add (Xb) and
// optional fused final combine:  Y = addPost + relu(ln)   (layer-1 tail).
// ---------------------------------------------------------------------------
__global__ __launch_bounds__(256)
void k_ln(const float* __restrict__ Xa, const float* __restrict__ Xb,
          const float* __restrict__ g, const float* __restrict__ bias,
          float* __restrict__ Y, const float* __restrict__ addPost) {
  __shared__ float red[256];
  const int row = blockIdx.x;
  const int t = threadIdx.x;
  float x = Xa[(size_t)row * DM + t];
  if (Xb) x += Xb[(size_t)row * DM + t];

  red[t] = x;
  __syncthreads();
  for (int off = 128; off; off >>= 1) {
    if (t < off) red[t] += red[t + off];
    __syncthreads();
  }
  float mu = red[0] * (1.0f / DM);
  __syncthreads();

  float dx = x - mu;
  red[t] = dx * dx;
  __syncthreads();
  for (int off = 128; off; off >>= 1) {
    if (t < off) red[t] += red[t + off];
    __syncthreads();
  }
  float var = red[0] * (1.0f / DM);

  float y = dx * rsqrtf(var + 1e-5f) * g[t] + bias[t];
  if (addPost) y = addPost[(size_t)row * DM + t] + fmaxf(y, 0.0f);
  Y[(size_t)row * DM + t] = y;
}

// ---------------------------------------------------------------------------
// Host-side orchestration.
// ---------------------------------------------------------------------------
extern "C" void kernel_launch(void* const* d_in, const int* in_sizes, int n_in,
                              void* d_out, int out_size, void* d_ws, size_t ws_size,
                              hipStream_t stream) {
  (void)in_sizes; (void)n_in; (void)out_size; (void)ws_size;

  const float* interact = (const float*)d_in[1];   // [B,S,2D]
  const float* next_q   = (const float*)d_in[2];   // [B,S,D]
  const float* lin_w    = (const float*)d_in[3];   // [2D,D]
  const float* lin_b    = (const float*)d_in[4];   // [D]
  const float* Wq       = (const float*)d_in[5];
  const float* bq       = (const float*)d_in[6];
  const float* Wk       = (const float*)d_in[7];
  const float* bk       = (const float*)d_in[8];
  const float* Wv       = (const float*)d_in[9];
  const float* bv       = (const float*)d_in[10];
  const float* ln1g     = (const float*)d_in[11];
  const float* ln1b     = (const float*)d_in[12];
  const float* w1       = (const float*)d_in[13];  // [L,D,2D]
  const float* b1       = (const float*)d_in[14];  // [L,2D]
  const float* w2       = (const float*)d_in[15];  // [L,2D,D]
  const float* b2       = (const float*)d_in[16];  // [L,D]
  const float* lng      = (const float*)d_in[17];
  const float* lnb      = (const float*)d_in[18];
  const float* pke      = (const float*)d_in[19];  // [512,32]
  const float* pve      = (const float*)d_in[20];  // [512,32]
  float* out = (float*)d_out;                      // [B,S,D]

  float* ws = (float*)d_ws;
  const size_t NT = (size_t)NROW * DM;             // 4,194,304 floats
  float* X    = ws;                                // layer-0 input x
  float* Qb   = ws + 1 * NT;                       // q, later reused as Z
  float* KTb  = ws + 2 * NT;                       // k^T; KTb..+2NT reused as H
  float* Vb   = ws + 3 * NT;                       // v
  float* Eb   = ws + 4 * NT;                       // attn-out, later Y
  float* O0   = ws + 5 * NT;                       // layer-0 output
  float* PKET = ws + 6 * NT;                       // [32,512]
  float* Hb   = KTb;                               // [16384,512]
  float* Zb   = Qb;                                // post-LN1 activations

  const int tilesMN = (NROW / 16) * (DM / 16);     // 16384 tiles for N=256

  k_transpose_pke<<<(SEQ * HD + 255) / 256, 256, 0, stream>>>(pke, PKET);
  k_lin_in<<<tilesMN, 32, 0, stream>>>(interact, lin_w, lin_b, X);

  for (int l = 0; l < 2; ++l) {
    const float* x = (l == 0) ? X : O0;
    k_qkv<<<dim3(tilesMN, 3), 32, 0, stream>>>(
        next_q, x,
        Wq + (size_t)l * DM * DM, bq + (size_t)l * DM,
        Wk + (size_t)l * DM * DM, bk + (size_t)l * DM,
        Wv + (size_t)l * DM * DM, bv + (size_t)l * DM,
        Qb, KTb, Vb);

    k_attn<<<32 * NH * (SEQ / 16), 32, 0, stream>>>(Qb, KTb, Vb, PKET, pve, Eb);

    // Z = LN(attn_out + query)
    k_ln<<<NROW, 256, 0, stream>>>(Eb, next_q,
                                   ln1g + (size_t)l * DM, ln1b + (size_t)l * DM,
                                   Zb, nullptr);

    // H = relu(Z @ W1 + b1)   [16384,256] @ [256,512]
    k_gemm<<<(NROW / 16) * (2 * DM / 16), 32, 0, stream>>>(
        Zb, DM, w1 + (size_t)l * DM * 2 * DM, 2 * DM,
        b1 + (size_t)l * 2 * DM, nullptr, Hb, 2 * DM, DM, 2 * DM / 16, 1);

    // Y = H @ W2 + b2 + Z     [16384,512] @ [512,256]
    k_gemm<<<tilesMN, 32, 0, stream>>>(
        Hb, 2 * DM, w2 + (size_t)l * 2 * DM * DM, DM,
        b2 + (size_t)l * DM, Zb, Eb, DM, 2 * DM, DM / 16, 0);

    if (l == 0) {
      k_ln<<<NROW, 256, 0, stream>>>(Eb, nullptr, lng, lnb, O0, nullptr);
    } else {
      // out = out0 + relu(LN(Y))  fused final combine
      k_ln<<<NROW, 256, 0, stream>>>(Eb, nullptr, lng + DM, lnb + DM, out, O0);
    }
  }
}